// GATv2_25555055411698
// MI455X (gfx1250) — compile-verified
//
#include <hip/hip_runtime.h>
#include <hip/hip_bf16.h>
#include <math.h>

// ---------------------------------------------------------------------------
// GATv2 (2 layers) for MI455X / gfx1250, wave32.
//  * Node transforms: V_WMMA_F32_16X16X4_F32, one wave per 16x64 row-block
//    (4 accumulators share one A fragment per k-step).
//  * Attention: CSR-by-destination + one wave per node, single-pass online
//    softmax. Each lane owns 2 of 64 channels -> coalesced 256B row gathers,
//    cross-lane logit reduction via __shfl_xor, no float atomics in hot loop.
// ---------------------------------------------------------------------------

typedef __attribute__((ext_vector_type(2))) float v2f;
typedef __attribute__((ext_vector_type(8))) float v8f;

#define NEG_SLOPE 0.2f

__device__ __forceinline__ float lrelu(float v) {
    return v > 0.0f ? v : NEG_SLOPE * v;
}

// ---------------------------------------------------------------------------
// generic fills
// ---------------------------------------------------------------------------
__global__ void fill_f32(float* __restrict__ p, float v, size_t n) {
    size_t i = (size_t)blockIdx.x * blockDim.x + threadIdx.x;
    size_t stride = (size_t)gridDim.x * blockDim.x;
    for (; i < n; i += stride) p[i] = v;
}
__global__ void fill_u32(unsigned* __restrict__ p, unsigned v, size_t n) {
    size_t i = (size_t)blockIdx.x * blockDim.x + threadIdx.x;
    size_t stride = (size_t)gridDim.x * blockDim.x;
    for (; i < n; i += stride) p[i] = v;
}

// ---------------------------------------------------------------------------
// per-destination counts (incl. self loop, pre-set to 1) and weight sums
// ---------------------------------------------------------------------------
__global__ void count_kernel(const int* __restrict__ dst,
                             const float* __restrict__ ew,
                             int* __restrict__ cnt,
                             float* __restrict__ wsum, int E) {
    int e = blockIdx.x * blockDim.x + threadIdx.x;
    if (e >= E) return;
    int d = dst[e];
    atomicAdd(&cnt[d], 1);
    atomicAdd(&wsum[d], ew[e]);
}

// self-loop weight = mean of incoming real-edge weights (PyG fill='mean')
__global__ void loop_attr_kernel(const int* __restrict__ cnt,
                                 const float* __restrict__ wsum,
                                 float* __restrict__ loopw, int Nn) {
    int i = blockIdx.x * blockDim.x + threadIdx.x;
    if (i >= Nn) return;
    float degR = (float)(cnt[i] - 1);  // real-edge in-degree
    loopw[i] = wsum[i] / fmaxf(degR, 1.0f);
}

// ---------------------------------------------------------------------------
// 3-phase exclusive scan of cnt[] -> rowptr[] (Nn+1 entries)
// ---------------------------------------------------------------------------
__global__ void scan_chunk(const int* __restrict__ cnt,
                           int* __restrict__ scanbuf,
                           int* __restrict__ blockSums, int n) {
    __shared__ int sm[256];
    int i = blockIdx.x * 256 + threadIdx.x;
    sm[threadIdx.x] = (i < n) ? cnt[i] : 0;
    __syncthreads();
#pragma unroll
    for (int off = 1; off < 256; off <<= 1) {
        int t = (threadIdx.x >= off) ? sm[threadIdx.x - off] : 0;
        __syncthreads();
        sm[threadIdx.x] += t;
        __syncthreads();
    }
    if (i < n) scanbuf[i] = sm[threadIdx.x];  // inclusive within chunk
    if (threadIdx.x == 255) blockSums[blockIdx.x] = sm[255];
}

__global__ void scan_block_sums(int* __restrict__ blockSums, int nb) {
    __shared__ int sm[512];
    int t = threadIdx.x;
    sm[t] = (t < nb) ? blockSums[t] : 0;
    __syncthreads();
#pragma unroll
    for (int off = 1; off < 512; off <<= 1) {
        int v = (t >= off) ? sm[t - off] : 0;
        __syncthreads();
        sm[t] += v;
        __syncthreads();
    }
    if (t < nb) blockSums[t] = (t == 0) ? 0 : sm[t - 1];  // exclusive
}

__global__ void scan_finish(const int* __restrict__ scanbuf,
                            const int* __restrict__ blockSums,
                            int* __restrict__ rowptr, int n) {
    int i = blockIdx.x * blockDim.x + threadIdx.x;
    if (i < n) rowptr[i + 1] = scanbuf[i] + blockSums[i >> 8];
    if (i == 0) rowptr[0] = 0;
}

__global__ void cursor_init(const int* __restrict__ rowptr,
                            int* __restrict__ cursor, int n) {
    int i = blockIdx.x * blockDim.x + threadIdx.x;
    if (i < n) cursor[i] = rowptr[i];
}

__global__ void scatter_real(const int* __restrict__ dst,
                             int* __restrict__ cursor,
                             int* __restrict__ csr, int E) {
    int e = blockIdx.x * blockDim.x + threadIdx.x;
    if (e >= E) return;
    int pos = atomicAdd(&cursor[dst[e]], 1);
    csr[pos] = e;
}

__global__ void scatter_self(int* __restrict__ cursor,
                             int* __restrict__ csr, int E, int Nn) {
    int i = blockIdx.x * blockDim.x + threadIdx.x;
    if (i >= Nn) return;
    int pos = atomicAdd(&cursor[i], 1);
    csr[pos] = E + i;  // encode self loop of node i
}

// ---------------------------------------------------------------------------
// C[M,64] = A[M,K] @ W[K,64] + bias   via V_WMMA_F32_16X16X4_F32
// One wave computes a full 16x64 row-block: 4 accumulators, one shared A
// fragment per k-step (4 independent WMMAs back-to-back). 8 waves / block.
//
// f32 WMMA fragment layouts (wave32):
//   A 16x4:  lanes 0-15 -> M=lane,    {v0,v1} = K = k0+0, k0+1
//            lanes 16-31 -> M=lane-16, {v0,v1} = K = k0+2, k0+3
//   B 4x16:  mirrored (N across lanes, K across vgpr+half)
//   C 16x16: lanes 0-15 N=lane, vgpr r -> M = r + 8*half
// ---------------------------------------------------------------------------
template <int K>
__global__ __launch_bounds__(256) void gemm_bias_wmma(
    const float* __restrict__ A, const float* __restrict__ W,
    const float* __restrict__ bias, float* __restrict__ C, int M) {
    int mt = blockIdx.x * (blockDim.x >> 5) + (threadIdx.x >> 5);
    if (mt >= (M >> 4)) return;

    int lane = threadIdx.x & 31;
    int hf = lane >> 4;
    int lr = lane & 15;

    const float* Arow = A + (size_t)(mt * 16 + lr) * K;
    const float* Wb = W + lr;  // B column base for nt*16 offset below

    v8f acc0 = {}, acc1 = {}, acc2 = {}, acc3 = {};
#pragma unroll 2
    for (int k0 = 0; k0 < K; k0 += 4) {
        int ka = k0 + 2 * hf;
        v2f a;
        a.x = Arow[ka];
        a.y = Arow[ka + 1];
        const float* W0 = Wb + (size_t)ka * 64;
        const float* W1 = Wb + (size_t)(ka + 1) * 64;
        v2f b0, b1, b2, b3;
        b0.x = W0[0];
        b0.y = W1[0];
        b1.x = W0[16];
        b1.y = W1[16];
        b2.x = W0[32];
        b2.y = W1[32];
        b3.x = W0[48];
        b3.y = W1[48];
        acc0 = __builtin_amdgcn_wmma_f32_16x16x4_f32(false, a, false, b0,
                                                     (short)0, acc0, false, false);
        acc1 = __builtin_amdgcn_wmma_f32_16x16x4_f32(false, a, false, b1,
                                                     (short)0, acc1, false, false);
        acc2 = __builtin_amdgcn_wmma_f32_16x16x4_f32(false, a, false, b2,
                                                     (short)0, acc2, false, false);
        acc3 = __builtin_amdgcn_wmma_f32_16x16x4_f32(false, a, false, b3,
                                                     (short)0, acc3, false, false);
    }

    float bs0 = bias[lr];
    float bs1 = bias[16 + lr];
    float bs2 = bias[32 + lr];
    float bs3 = bias[48 + lr];
    float* Crow = C + (size_t)(mt * 16 + 8 * hf) * 64 + lr;
#pragma unroll
    for (int r = 0; r < 8; ++r) {
        float* p = Crow + (size_t)r * 64;
        p[0] = acc0[r] + bs0;
        p[16] = acc1[r] + bs1;
        p[32] = acc2[r] + bs2;
        p[48] = acc3[r] + bs3;
    }
}

// ---------------------------------------------------------------------------
// One wave32 per destination node; single-pass online-softmax aggregation.
// Lane owns channels (2*lane, 2*lane+1); for H=2 lanes 0-15 are head 0 and
// lanes 16-31 head 1, so the logit reduction stays within a half-wave.
// ---------------------------------------------------------------------------
template <int H, bool RELU>
__global__ __launch_bounds__(256) void node_attn(
    const int* __restrict__ csr, const int* __restrict__ rowptr,
    const int* __restrict__ src, const float* __restrict__ ew,
    const float* __restrict__ loopw, const float* __restrict__ xl,
    const float* __restrict__ xr, const float* __restrict__ We,
    const float* __restrict__ att, const float* __restrict__ bias,
    float* __restrict__ out, int E, int Nn) {
    int wave = blockIdx.x * (blockDim.x >> 5) + (threadIdx.x >> 5);
    if (wave >= Nn) return;
    const int n = wave;
    const int lane = threadIdx.x & 31;
    const int c0 = lane * 2;

    // per-lane constants for its 2 channels
    float2 xrv = *(const float2*)(xr + (size_t)n * 64 + c0);
    float2 wev = *(const float2*)(We + c0);
    float2 atv = *(const float2*)(att + c0);

    float m = -INFINITY, s = 0.0f, acc0 = 0.0f, acc1 = 0.0f;

    const int beg = rowptr[n], end = rowptr[n + 1];
    for (int idx = beg; idx < end; ++idx) {
        int eid = csr[idx];
        int sN;
        float w;
        if (eid < E) {
            sN = src[eid];
            w = ew[eid];
        } else {
            sN = n;  // self loop
            w = loopw[n];
        }
        float2 xlv = *(const float2*)(xl + (size_t)sN * 64 + c0);  // 256B/wave

        float v0 = lrelu(xlv.x + xrv.x + w * wev.x);
        float v1 = lrelu(xlv.y + xrv.y + w * wev.y);
        float partial = atv.x * v0 + atv.y * v1;

        // reduce over this head's lanes (16 for H=2, 32 for H=1)
        const int RED = (H == 2) ? 8 : 16;
#pragma unroll
        for (int off = 1; off <= RED; off <<= 1)
            partial += __shfl_xor(partial, off, 32);
        float logit = partial;

        // online softmax update (every node has >=1 edge: its self loop)
        float mnew = fmaxf(m, logit);
        float scale = __expf(m - mnew);
        float pe = __expf(logit - mnew);
        s = s * scale + pe;
        acc0 = acc0 * scale + pe * xlv.x;
        acc1 = acc1 * scale + pe * xlv.y;
        m = mnew;
    }

    float inv = 1.0f / s;
    float2 bv = *(const float2*)(bias + c0);
    float o0 = acc0 * inv + bv.x;
    float o1 = acc1 * inv + bv.y;
    if (RELU) {
        o0 = fmaxf(o0, 0.0f);
        o1 = fmaxf(o1, 0.0f);
    }
    float2 res;
    res.x = o0;
    res.y = o1;
    *(float2*)(out + (size_t)n * 64 + c0) = res;
}

// ---------------------------------------------------------------------------
// host orchestration
// ---------------------------------------------------------------------------
extern "C" void kernel_launch(void* const* d_in, const int* in_sizes, int n_in,
                              void* d_out, int out_size, void* d_ws,
                              size_t ws_size, hipStream_t stream) {
    const float* x    = (const float*)d_in[0];
    const int*   ei   = (const int*)d_in[1];
    const float* ew   = (const float*)d_in[2];
    const float* W1l  = (const float*)d_in[3];
    const float* b1l  = (const float*)d_in[4];
    const float* W1r  = (const float*)d_in[5];
    const float* b1r  = (const float*)d_in[6];
    const float* We1  = (const float*)d_in[7];
    const float* att1 = (const float*)d_in[8];
    const float* bi1  = (const float*)d_in[9];
    const float* W2l  = (const float*)d_in[10];
    const float* b2l  = (const float*)d_in[11];
    const float* W2r  = (const float*)d_in[12];
    const float* b2r  = (const float*)d_in[13];
    const float* We2  = (const float*)d_in[14];
    const float* att2 = (const float*)d_in[15];
    const float* bi2  = (const float*)d_in[16];

    const int Nn = in_sizes[0] / 128;  // 100000
    const int E  = in_sizes[1] / 2;    // 1000000
    const int Et = E + Nn;
    const int* srcp = ei;
    const int* dstp = ei + E;

    // workspace layout
    float* ws     = (float*)d_ws;
    float* wsum   = ws;                          // [Nn] f32
    float* loopw  = wsum + Nn;                   // [Nn] f32
    int*   cnt    = (int*)(loopw + Nn);          // [Nn]
    int*   rowptr = cnt + Nn;                    // [Nn+1]
    int*   cursor = rowptr + (Nn + 1);           // [Nn]
    int*   scanb  = cursor + Nn;                 // [Nn]
    int*   bsums  = scanb + Nn;                  // [<=512]
    int*   csr    = bsums + 512;                 // [Et]
    float* bufA   = (float*)(csr + Et);          // [Nn*64] xl
    float* bufB   = bufA + (size_t)Nn * 64;      // [Nn*64] xr
    float* bufC   = bufB + (size_t)Nn * 64;      // [Nn*64] h
    float* out    = (float*)d_out;               // [Nn*64]

    const int TB = 256;
    const int gEr = (E + TB - 1) / TB;
    const int gN  = (Nn + TB - 1) / TB;
    const int nbScan = (Nn + 255) / 256;  // 391 (<512)
    const int gTiles = ((Nn / 16) + 7) / 8;  // one wave per 16x64 row-block
    const int gNodes = (Nn + 7) / 8;         // 8 waves / block

    // ---- CSR build + self-loop mean weights ----
    fill_f32<<<1024, TB, 0, stream>>>(wsum, 0.0f, (size_t)Nn);
    fill_u32<<<1024, TB, 0, stream>>>((unsigned*)cnt, 1u, (size_t)Nn);  // +self
    count_kernel<<<gEr, TB, 0, stream>>>(dstp, ew, cnt, wsum, E);
    loop_attr_kernel<<<gN, TB, 0, stream>>>(cnt, wsum, loopw, Nn);
    scan_chunk<<<nbScan, 256, 0, stream>>>(cnt, scanb, bsums, Nn);
    scan_block_sums<<<1, 512, 0, stream>>>(bsums, nbScan);
    scan_finish<<<gN, TB, 0, stream>>>(scanb, bsums, rowptr, Nn);
    cursor_init<<<gN, TB, 0, stream>>>(rowptr, cursor, Nn);
    scatter_real<<<gEr, TB, 0, stream>>>(dstp, cursor, csr, E);
    scatter_self<<<gN, TB, 0, stream>>>(cursor, csr, E, Nn);

    // ---- layer 1 ----
    gemm_bias_wmma<128><<<gTiles, TB, 0, stream>>>(x, W1l, b1l, bufA, Nn);
    gemm_bias_wmma<128><<<gTiles, TB, 0, stream>>>(x, W1r, b1r, bufB, Nn);
    node_attn<2, true><<<gNodes, TB, 0, stream>>>(
        csr, rowptr, srcp, ew, loopw, bufA, bufB, We1, att1, bi1, bufC, E, Nn);

    // ---- layer 2 ----
    gemm_bias_wmma<64><<<gTiles, TB, 0, stream>>>(bufC, W2l, b2l, bufA, Nn);
    gemm_bias_wmma<64><<<gTiles, TB, 0, stream>>>(bufC, W2r, b2r, bufB, Nn);
    node_attn<1, false><<<gNodes, TB, 0, stream>>>(
        csr, rowptr, srcp, ew, loopw, bufA, bufB, We2, att2, bi2, out, E, Nn);
}